// MaskedWeight_31353261261544
// MI455X (gfx1250) — compile-verified
//
#include <hip/hip_runtime.h>
#include <math.h>

typedef float v2f __attribute__((ext_vector_type(2)));
typedef float v8f __attribute__((ext_vector_type(8)));

#define FDIM 2048
#define BATCH 1024
#define NDIM 64          // DIM in reference
#define BLK 32           // opd = ipd = FDIM/NDIM
#define MSUB 4           // M sub-tiles (16 rows each) per wave in GEMM

// ---------------------------------------------------------------------------
// Stage 1: per-row weight preprocessing.
//   w[r,c]   = exp(weight) on diag 32x32 block, weight for c < br*32, 0 above
//   wsn[r]   = sum_c w^2
//   w_n[r,c] = exp(diag_weight[r]) * w[r,c] * rsqrt(wsn[r])   (written to ws)
//   blocksT[d][k][o] = diag_weight[r] + weight[r, d*32+k] - 0.5*log(wsn[r])
//                      with r = d*32+o   (transposed for coalesced logdet reads)
// One 256-thread block per row; row values kept in registers (<= 8 per thread).
// ---------------------------------------------------------------------------
__global__ __launch_bounds__(256)
void prep_kernel(const float* __restrict__ weight,
                 const float* __restrict__ diag_weight,
                 float* __restrict__ w_n,      // [FDIM][FDIM] (cols < (br+1)*32)
                 float* __restrict__ blocksT)  // [NDIM][32][32]  (d, k, o)
{
    const int r     = blockIdx.x;
    const int br    = r >> 5;              // block row
    const int ncols = (br + 1) * BLK;      // nonzero columns in this row
    const int t     = threadIdx.x;
    const float* wrow = weight + (size_t)r * FDIM;

    float vals[8];
    int   cnt = 0;
    float ss  = 0.f;
    for (int c = t; c < ncols; c += 256) {
        float raw = wrow[c];
        float wv  = (c >= br * BLK) ? __expf(raw) : raw;
        vals[cnt++] = wv;
        ss += wv * wv;
    }

    // wave32 shuffle reduce, then cross-wave via LDS
    #pragma unroll
    for (int off = 16; off > 0; off >>= 1)
        ss += __shfl_down(ss, off, 32);

    __shared__ float red[8];
    const int wave = t >> 5, lane = t & 31;
    if (lane == 0) red[wave] = ss;
    __syncthreads();
    float wsn = red[0] + red[1] + red[2] + red[3] +
                red[4] + red[5] + red[6] + red[7];

    const float dw    = diag_weight[r];
    const float scale = __expf(dw) * rsqrtf(wsn);

    float* orow = w_n + (size_t)r * FDIM;
    cnt = 0;
    for (int c = t; c < ncols; c += 256)
        orow[c] = vals[cnt++] * scale;

    if (t < BLK) {
        const int o = r & 31;
        const float bval = dw + wrow[br * BLK + t] - 0.5f * __logf(wsn);
        blocksT[(size_t)br * (BLK * BLK) + t * BLK + o] = bval;
    }
}

// ---------------------------------------------------------------------------
// Stage 2: out = inputs @ w_n.T + bias  via V_WMMA_F32_16X16X4_F32.
// One wave computes a (16*MSUB) x 16 output tile (MSUB=4 -> 64x16). w_n is
// block lower-triangular: rows of a 16-wide N tile all live in block-row
// br = n0/32, so the K loop stops at (br+1)*32 (halves average wmma count).
//
// Tile indices are forced into SGPRs with readfirstlane so the K-loop bound
// is scalar (s_cbranch loop, no exec masking) and can be unrolled; klim is
// always a multiple of 32, so a K-step of 16 needs no remainder loop.
// MSUB=4 gives 4 independent accumulator chains (hides XDL latency) and 4x
// reuse of each B fragment.
//
// fp32 WMMA fragment layout (ISA 7.12.2):
//   A 16x4 : lanes 0-15 hold (K=0,K=1) of row M=lane, lanes 16-31 hold (K=2,K=3)
//   B 4x16 : lanes 0-15 hold (K=0,K=1) of col N=lane, lanes 16-31 hold (K=2,K=3)
//   C/D    : VGPR j -> row j (lanes 0-15) / row j+8 (lanes 16-31), lane -> col
// Both fragment loads are per-lane contiguous float2 (global_load_b64).
// ---------------------------------------------------------------------------
__global__ __launch_bounds__(256)
void gemm_kernel(const float* __restrict__ inputs,  // [BATCH][FDIM]
                 const float* __restrict__ w_n,     // [FDIM][FDIM]
                 const float* __restrict__ bias,    // [FDIM]
                 float* __restrict__ out)           // [BATCH][FDIM]
{
    const int tid  = threadIdx.x;
    const int wave = tid >> 5;
    const int lane = tid & 31;
    const int gw   = blockIdx.x * 8 + wave;

    const int ntilesN = FDIM / 16;               // 128
    // Force wave-uniform tile coordinates into SGPRs -> scalar loop control.
    const int tm = __builtin_amdgcn_readfirstlane(gw / ntilesN);
    const int tn = __builtin_amdgcn_readfirstlane(gw % ntilesN);
    const int m0 = tm * 16 * MSUB;
    const int n0 = tn * 16;
    const int klim = ((n0 >> 5) + 1) * BLK;      // triangular K limit (mult of 32)

    const int lrow  = lane & 15;
    const int khalf = (lane >> 4) * 2;           // 0 for lanes 0-15, 2 for hi

    const float* arow[MSUB];
    #pragma unroll
    for (int s = 0; s < MSUB; s++)
        arow[s] = inputs + (size_t)(m0 + s * 16 + lrow) * FDIM + khalf;
    const float* brow = w_n + (size_t)(n0 + lrow) * FDIM + khalf;

    v8f acc[MSUB];
    #pragma unroll
    for (int s = 0; s < MSUB; s++)
        acc[s] = (v8f){0.f, 0.f, 0.f, 0.f, 0.f, 0.f, 0.f, 0.f};

    // K-step of 16 = 4 wmma steps; klim % 16 == 0 always.
    for (int k0 = 0; k0 < klim; k0 += 16) {
        v2f bf[4];
        #pragma unroll
        for (int q = 0; q < 4; q++)
            bf[q] = *(const v2f*)(brow + k0 + 4 * q);

        v2f af[MSUB][4];
        #pragma unroll
        for (int s = 0; s < MSUB; s++)
            #pragma unroll
            for (int q = 0; q < 4; q++)
                af[s][q] = *(const v2f*)(arow[s] + k0 + 4 * q);

        #pragma unroll
        for (int q = 0; q < 4; q++)
            #pragma unroll
            for (int s = 0; s < MSUB; s++)
                acc[s] = __builtin_amdgcn_wmma_f32_16x16x4_f32(
                    /*neg_a=*/false, af[s][q], /*neg_b=*/false, bf[q],
                    /*c_mod=*/(short)0, acc[s],
                    /*reuse_a=*/false, /*reuse_b=*/false);
    }

    const float bv    = bias[n0 + lrow];
    const int   colg  = n0 + lrow;
    const int   rbase = (lane >> 4) * 8;
    #pragma unroll
    for (int s = 0; s < MSUB; s++) {
        #pragma unroll
        for (int j = 0; j < 8; j++) {
            const int row = m0 + s * 16 + rbase + j;
            out[(size_t)row * FDIM + colg] = acc[s][j] + bv;
        }
    }
}

// ---------------------------------------------------------------------------
// Stage 3: logdet[b,d,o] = logsumexp_k( blocks[d,o,k] + grad[b,d,k] ), k<32.
// One wave per (b,d); lane = o. grad row broadcast across the wave via shfl,
// blocksT read lane-coalesced.
// ---------------------------------------------------------------------------
__global__ __launch_bounds__(256)
void logdet_kernel(const float* __restrict__ grad,     // [BATCH][NDIM][32][1]
                   const float* __restrict__ blocksT,  // [NDIM][32(k)][32(o)]
                   float* __restrict__ logdet)         // [BATCH][NDIM][32]
{
    const int tid  = threadIdx.x;
    const int wave = tid >> 5;
    const int lane = tid & 31;
    const int gw   = blockIdx.x * 8 + wave;   // (b,d) pair index
    const int d    = gw % NDIM;

    const float g = grad[(size_t)gw * BLK + lane];
    const float* bt = blocksT + (size_t)d * (BLK * BLK);

    float vals[32];
    float mx = -3.402823466e38f;
    #pragma unroll
    for (int k = 0; k < BLK; k++) {
        float v = bt[k * BLK + lane] + __shfl(g, k, 32);
        vals[k] = v;
        mx = fmaxf(mx, v);
    }
    float s = 0.f;
    #pragma unroll
    for (int k = 0; k < BLK; k++)
        s += __expf(vals[k] - mx);

    logdet[(size_t)gw * BLK + lane] = mx + __logf(s);
}

// ---------------------------------------------------------------------------
extern "C" void kernel_launch(void* const* d_in, const int* in_sizes, int n_in,
                              void* d_out, int out_size, void* d_ws, size_t ws_size,
                              hipStream_t stream) {
    const float* inputs      = (const float*)d_in[0];  // [1024,2048]
    const float* grad        = (const float*)d_in[1];  // [1024,64,32,1]
    const float* weight      = (const float*)d_in[2];  // [2048,2048]
    const float* diag_weight = (const float*)d_in[3];  // [2048,1]
    const float* bias        = (const float*)d_in[4];  // [2048]

    float* out    = (float*)d_out;                       // BATCH*FDIM
    float* logdet = out + (size_t)BATCH * FDIM;          // BATCH*NDIM*32

    float* w_n     = (float*)d_ws;                       // FDIM*FDIM floats
    float* blocksT = w_n + (size_t)FDIM * FDIM;          // NDIM*32*32 floats

    prep_kernel<<<FDIM, 256, 0, stream>>>(weight, diag_weight, w_n, blocksT);

    const int gemm_waves = (BATCH / (16 * MSUB)) * (FDIM / 16);  // 2048
    gemm_kernel<<<gemm_waves / 8, 256, 0, stream>>>(inputs, w_n, bias, out);

    logdet_kernel<<<(BATCH * NDIM) / 8, 256, 0, stream>>>(grad, blocksT, logdet);
}